// LeakyTopKActivation_21784074126076
// MI455X (gfx1250) — compile-verified
//
#include <hip/hip_runtime.h>
#include <hip/hip_bf16.h>
#include <stdint.h>

// LeakyTopKActivation on MI455X (gfx1250).
// out[b,n] = x[b,n] * 3.0 * (x[b,n] in row-top-614 ? 1.0 : 0.1)
//
// Memory-bound: 256 MiB traffic -> ~11us floor @ 23.3 TB/s.
// One workgroup (256 thr = 8 wave32) per row:
//   - CDNA5 async global->LDS bulk copy of the 16KB row (ASYNCcnt path)
//   - exact k-th-largest via 4-pass radix select; per-pass digit selection
//     done entirely inside one wave32 with __shfl_down (no workgroup barriers)
//   - registers hold the row slice, single coalesced store out.

#define ROW_N   4096
#define TOPK    614          // max(int(4096*0.15),1)
#define NTHR    256
#define EPT     (ROW_N / NTHR)   // 16 elements per thread
#define GAINF   3.0f
#define LEAKG   0.3f         // GAIN * LEAK

__global__ __launch_bounds__(NTHR)
void leaky_topk_kernel(const float* __restrict__ x, float* __restrict__ out) {
    __shared__ __align__(16) unsigned keys[ROW_N];   // async-DMA staging of the row
    __shared__ unsigned hist[256];                   // radix histogram
    __shared__ unsigned sh_prefix;
    __shared__ unsigned sh_want;

    const int t   = threadIdx.x;
    const int row = blockIdx.x;
    const float* rowp = x + (size_t)row * ROW_N;

    // Low 32 bits of a flat-cast LDS pointer are the LDS byte offset.
    unsigned ldsbase = (unsigned)(unsigned long long)(&keys[0]);

    // ---- CDNA5 async global -> LDS copy: 4 x b128 per thread = 16KB/row ----
    #pragma unroll
    for (int r = 0; r < 4; ++r) {
        unsigned byteoff = (unsigned)(r * (NTHR * 16) + t * 16);
        unsigned lds_off = ldsbase + byteoff;
        asm volatile("global_load_async_to_lds_b128 %0, %1, %2"
                     :: "v"(lds_off), "v"(byteoff), "s"(rowp)
                     : "memory");
    }
    asm volatile("s_wait_asynccnt 0" ::: "memory");
    __syncthreads();

    // ---- pull slice into registers; build order-preserving uint keys ----
    float    fv[EPT];
    unsigned kv[EPT];
    #pragma unroll
    for (int i = 0; i < EPT; ++i) {
        unsigned u = keys[t + i * NTHR];
        fv[i] = __uint_as_float(u);
        unsigned m = (unsigned)(((int)u) >> 31) | 0x80000000u;  // flip for order
        kv[i] = u ^ m;
    }

    // ---- exact radix select of the TOPK-th largest key (MSB -> LSB) ----
    unsigned prefix = 0u;
    unsigned want   = TOPK;
    #pragma unroll
    for (int pass = 0; pass < 4; ++pass) {
        const int shift = 24 - 8 * pass;

        hist[t] = 0u;
        __syncthreads();

        #pragma unroll
        for (int i = 0; i < EPT; ++i) {
            unsigned u = kv[i];
            bool ok = (pass == 0) || ((u >> (shift + 8)) == prefix);
            if (ok) atomicAdd(&hist[(u >> shift) & 255u], 1u);   // ds_add
        }
        __syncthreads();

        // Digit selection entirely inside wave 0: lane l owns bins [8l, 8l+8).
        if (t < 32) {
            unsigned part[8];
            unsigned seg = 0u;
            #pragma unroll
            for (int j = 0; j < 8; ++j) {
                part[j] = hist[t * 8 + j];
                seg += part[j];
            }
            // 5-step inclusive suffix scan across 32 lanes (wave32, no barriers):
            // suf = sum of seg over lanes >= this lane
            unsigned suf = seg;
            #pragma unroll
            for (int off = 1; off < 32; off <<= 1) {
                unsigned v = (unsigned)__shfl_down((int)suf, (unsigned)off, 32);
                if (t + off < 32) suf += v;
            }
            unsigned cg = suf - seg;   // count in bins above this lane's segment
            // Walk own 8 bins from high to low; unique global crossing bin wins.
            #pragma unroll
            for (int j = 7; j >= 0; --j) {
                unsigned cgb = cg + part[j];
                if (cgb >= want && cg < want) {      // crossing bin
                    sh_prefix = (prefix << 8) | (unsigned)(t * 8 + j);
                    sh_want   = want - cg;
                }
                cg = cgb;
            }
        }
        __syncthreads();
        prefix = sh_prefix;
        want   = sh_want;
    }
    const unsigned thresh = prefix;   // exact bit pattern of k-th largest

    // ---- apply leaky mask from registers, coalesced streaming store ----
    float* orow = out + (size_t)row * ROW_N;
    #pragma unroll
    for (int i = 0; i < EPT; ++i) {
        float scale = (kv[i] >= thresh) ? GAINF : LEAKG;
        orow[t + i * NTHR] = fv[i] * scale;
    }
}

extern "C" void kernel_launch(void* const* d_in, const int* in_sizes, int n_in,
                              void* d_out, int out_size, void* d_ws, size_t ws_size,
                              hipStream_t stream) {
    (void)n_in; (void)d_ws; (void)ws_size; (void)out_size;
    const float* x = (const float*)d_in[0];
    float* out = (float*)d_out;
    const int rows = in_sizes[0] / ROW_N;   // 8192
    leaky_topk_kernel<<<dim3(rows), dim3(NTHR), 0, stream>>>(x, out);
}